// AI4DEM_72842645340563
// MI455X (gfx1250) — compile-verified
//
#include <hip/hip_runtime.h>
#include <math.h>

// ---------------- problem geometry ----------------
#define ZDIM 128
#define YDIM 128
#define XDIM 128
#define NCELL (ZDIM * YDIM * XDIM)

// tile: 16x4x4 cells per block, 256 threads (8 waves, wave32)
#define TX 16
#define TY 4
#define TZ 4
#define HXE (TX + 4) // 20
#define HYE (TY + 4) // 8
#define HZE (TZ + 4) // 8
#define HCELL (HXE * HYE * HZE) // 1280
#define NTHR (TX * TY * TZ)     // 256

#define D_CELL 0.003f
#define TWO_D (2.0f * D_CELL)
#define KN_ 10000.0f
#define MU_ 0.43f
#define EPS_ 1e-4f

typedef __attribute__((ext_vector_type(4))) unsigned int v4u;
typedef __attribute__((ext_vector_type(8))) int v8i;
typedef __attribute__((ext_vector_type(4))) int v4i;

__global__ __launch_bounds__(NTHR) void dem_stencil_kernel(
    const float* __restrict__ xg,  const float* __restrict__ yg,
    const float* __restrict__ zg,  const float* __restrict__ vxg,
    const float* __restrict__ vyg, const float* __restrict__ vzg,
    const float* __restrict__ wxg, const float* __restrict__ wyg,
    const float* __restrict__ wzg, const float* __restrict__ maskg,
    float* __restrict__ out, float eta)
{
  __shared__ float sf[9][HCELL];

  const int tid = threadIdx.x;
  const int x0 = (int)blockIdx.x * TX - 2;
  const int y0 = (int)blockIdx.y * TY - 2;
  const int z0 = (int)blockIdx.z * TZ - 2;

  const float* fields[9] = {xg, yg, zg, vxg, vyg, vzg, wxg, wyg, wzg};

  const bool interior = (x0 >= 0) && (x0 + HXE <= XDIM) &&
                        (y0 >= 0) && (y0 + HYE <= YDIM) &&
                        (z0 >= 0) && (z0 + HZE <= ZDIM);
  if (interior) {
    if (tid < 32) { // wave 0 drives the Tensor Data Mover (EXEC-independent DMA)
      // ---- D# group 1 (shared by all 9 fields), per ISA 8.4 ----
      v8i g1;
      g1[0] = (2 << 16);                           // data_size = 4 bytes
      g1[1] = (int)((XDIM & 0xFFFF) << 16);        // tensor_dim0[15:0]
      g1[2] = (int)((XDIM >> 16) | ((YDIM & 0xFFFF) << 16)); // dim0 hi | dim1 lo
      g1[3] = (int)((YDIM >> 16) | (HXE << 16));   // dim1 hi | tile_dim0
      g1[4] = (int)(HYE | (HZE << 16));            // tile_dim1 | tile_dim2
      g1[5] = (int)XDIM;                           // tensor_dim0_stride[31:0]
      g1[6] = (int)(((XDIM * YDIM) & 0xFFFF) << 16); // stride0 hi16 | stride1 lo16
      g1[7] = (int)((XDIM * YDIM) >> 16);          // stride1[47:16]
      // ---- D# group 2 / 3, per ISA 8.5 / 8.6 (3D tile, no iterate) ----
      v4i g2; g2[0] = ZDIM; g2[1] = 0; g2[2] = 0; g2[3] = 0;
      v4i g3; g3[0] = 0; g3[1] = 0; g3[2] = 0; g3[3] = 0;

      const unsigned long long tile_off =
          (unsigned long long)(((z0 * YDIM + y0) * XDIM + x0)) * 4ull;
#pragma unroll
      for (int f = 0; f < 9; ++f) {
        const unsigned long long ga =
            (unsigned long long)fields[f] + tile_off;
        const unsigned lds = (unsigned)(unsigned long long)(&sf[f][0]);
        v4u g0;
        g0[0] = 1u;                                 // count = 1 descriptor
        g0[1] = lds;                                // lds_addr (bytes)
        g0[2] = (unsigned)(ga & 0xFFFFFFFFull);     // global_addr[31:0]
        g0[3] = (unsigned)((ga >> 32) & 0x1FFFFFFu) | (2u << 30); // type = 2
        // Portable TDM issue (bypasses clang builtin arity differences)
        asm volatile("tensor_load_to_lds %0, %1, %2, %3"
                     :
                     : "s"(g0), "s"(g1), "s"(g2), "s"(g3)
                     : "memory");
      }
      asm volatile("s_wait_tensorcnt 0x0" ::: "memory");
    }
  } else { // periodic-wrap staging for boundary blocks
    for (int c = tid; c < HCELL; c += NTHR) {
      const int hx = c % HXE;
      const int t1 = c / HXE;
      const int hy = t1 % HYE;
      const int hz = t1 / HYE;
      const int gx = (x0 + hx) & (XDIM - 1);
      const int gy = (y0 + hy) & (YDIM - 1);
      const int gz = (z0 + hz) & (ZDIM - 1);
      const int g = (gz * YDIM + gy) * XDIM + gx;
#pragma unroll
      for (int f = 0; f < 9; ++f) sf[f][c] = fields[f][g];
    }
  }
  __syncthreads();

  // ---------------- per-cell compute ----------------
  const int lx = tid % TX;
  const int tt = tid / TX;
  const int ly = tt % TY;
  const int lz = tt / TY;
  const int hx = lx + 2, hy = ly + 2, hz = lz + 2;
  const int coff = (hz * HYE + hy) * HXE + hx;

  const float cx  = sf[0][coff], cy  = sf[1][coff], cz  = sf[2][coff];
  const float cvx = sf[3][coff], cvy = sf[4][coff], cvz = sf[5][coff];
  const float cwx = sf[6][coff], cwy = sf[7][coff], cwz = sf[8][coff];

  const int gx = x0 + hx, gy = y0 + hy, gz = z0 + hz; // centers always in-range
  const int gidx = (gz * YDIM + gy) * XDIM + gx;
  // mask is read exactly once per cell -> non-temporal, keep L2 for the fields
  const float m = __builtin_nontemporal_load(&maskg[gidx]);

  float fxc = 0.f, fyc = 0.f, fzc = 0.f;
  float fxd = 0.f, fyd = 0.f, fzd = 0.f;
  float fxf = 0.f, fyf = 0.f, fzf = 0.f;
  float tqx = 0.f, tqy = 0.f, tqz = 0.f;

#pragma unroll 1
  for (int sxo = -2; sxo <= 2; ++sxo) {
#pragma unroll 1
    for (int syo = -2; syo <= 2; ++syo) {
#pragma unroll
      for (int szo = -2; szo <= 2; ++szo) {
        const int noff = ((hz - szo) * HYE + (hy - syo)) * HXE + (hx - sxo);

        const float dx = cx - sf[0][noff];
        const float dy = cy - sf[1][noff];
        const float dz = cz - sf[2][noff];
        const float d2s = dx * dx + dy * dy + dz * dz;
        const float dist = __fsqrt_rn(d2s);
        const float inv = __frcp_rn(fmaxf(EPS_, dist));
        const bool contact = dist < TWO_D;

        const float fn_col = KN_ * (dist - TWO_D);
        fxc += contact ? fn_col * dx * inv : 0.0f;
        fyc += contact ? fn_col * dy * inv : 0.0f;
        fzc += contact ? fn_col * dz * inv : 0.0f;
        const float fn_col_m = contact ? fn_col : 0.0f;

        const float swx = cwx + sf[6][noff];
        const float swy = cwy + sf[7][noff];
        const float swz = cwz + sf[8][noff];
        const float max_ = D_CELL * dx * inv;
        const float may_ = D_CELL * dy * inv;
        const float maz_ = D_CELL * dz * inv;
        const float vx_ang = (swy * maz_ - swz * may_) * m;
        const float vy_ang = (swz * max_ - swx * maz_) * m;
        const float vz_ang = (swx * may_ - swy * max_) * m;

        const float dvx = cvx - sf[3][noff];
        const float dvy = cvy - sf[4][noff];
        const float dvz = cvz - sf[5][noff];
        const float dv_n = (dvx * dx + dvy * dy + dvz * dz) * inv;
        const float fn_damp = contact ? -eta * dv_n : 0.0f;
        fxd += contact ? eta * dv_n * dx * inv : 0.0f;
        fyd += contact ? eta * dv_n * dy * inv : 0.0f;
        fzd += contact ? eta * dv_n * dz * inv : 0.0f;

        const float fn = fabsf(fn_col_m + fn_damp);
        const float inv2 = __frcp_rn(fmaxf(EPS_, dist * dist));
        const float vtx = dvx * (dy * dy + dz * dz) * inv2 + vx_ang;
        const float vty = dvy * (dx * dx + dz * dz) * inv2 + vy_ang;
        const float vtz = dvz * (dx * dx + dy * dy) * inv2 + vz_ang;
        const float vt = __fsqrt_rn(vtx * vtx + vty * vty + vtz * vtz);
        const float inv_vt = __frcp_rn(fmaxf(EPS_, vt));

        const float ffx = contact ? -(MU_ * fn * vtx * inv_vt) : 0.0f;
        const float ffy = contact ? -(MU_ * fn * vty * inv_vt) : 0.0f;
        const float ffz = contact ? -(MU_ * fn * vtz * inv_vt) : 0.0f;
        fxf += ffx;
        fyf += ffy;
        fzf += ffz;

        tqx += may_ * ffz - maz_ * ffy;
        tqy += maz_ * ffx - max_ * ffz;
        tqz += max_ * ffy - may_ * ffx;
      }
    }
  }

  // outputs are write-once streams -> non-temporal stores, don't evict inputs
  const size_t idx = (size_t)gidx;
  __builtin_nontemporal_store(fxc, &out[0u * (size_t)NCELL + idx]);
  __builtin_nontemporal_store(fyc, &out[1u * (size_t)NCELL + idx]);
  __builtin_nontemporal_store(fzc, &out[2u * (size_t)NCELL + idx]);
  __builtin_nontemporal_store(fxd, &out[3u * (size_t)NCELL + idx]);
  __builtin_nontemporal_store(fyd, &out[4u * (size_t)NCELL + idx]);
  __builtin_nontemporal_store(fzd, &out[5u * (size_t)NCELL + idx]);
  __builtin_nontemporal_store(fxf, &out[6u * (size_t)NCELL + idx]);
  __builtin_nontemporal_store(fyf, &out[7u * (size_t)NCELL + idx]);
  __builtin_nontemporal_store(fzf, &out[8u * (size_t)NCELL + idx]);
  __builtin_nontemporal_store(tqx, &out[9u * (size_t)NCELL + idx]);
  __builtin_nontemporal_store(tqy, &out[10u * (size_t)NCELL + idx]);
  __builtin_nontemporal_store(tqz, &out[11u * (size_t)NCELL + idx]);
}

extern "C" void kernel_launch(void* const* d_in, const int* in_sizes, int n_in,
                              void* d_out, int out_size, void* d_ws, size_t ws_size,
                              hipStream_t stream) {
  (void)in_sizes; (void)n_in; (void)out_size; (void)d_ws; (void)ws_size;

  const float* xg    = (const float*)d_in[0];
  const float* yg    = (const float*)d_in[1];
  const float* zg    = (const float*)d_in[2];
  const float* vxg   = (const float*)d_in[3];
  const float* vyg   = (const float*)d_in[4];
  const float* vzg   = (const float*)d_in[5];
  const float* wxg   = (const float*)d_in[6];
  const float* wyg   = (const float*)d_in[7];
  const float* wzg   = (const float*)d_in[8];
  const float* maskg = (const float*)d_in[9];
  float* out = (float*)d_out;

  // ETA computed in double precision exactly like the Python module
  const double alpha = -log(0.79) / 3.14159265358979323846;
  const double gma = alpha / sqrt(alpha * alpha + 1.0);
  const double mass = 4.0 / 3.0 * 3.1415926 * 0.003 * 0.003 * 0.003 * 674.0;
  const float eta = (float)(2.0 * gma * sqrt(10000.0 * mass / 2.0));

  dim3 grid(XDIM / TX, YDIM / TY, ZDIM / TZ); // (8, 32, 32)
  dim3 block(NTHR);
  dem_stencil_kernel<<<grid, block, 0, stream>>>(
      xg, yg, zg, vxg, vyg, vzg, wxg, wyg, wzg, maskg, out, eta);
}